// GaussianAdapter_82841329205488
// MI455X (gfx1250) — compile-verified
//
#include <hip/hip_runtime.h>
#include <hip/hip_bf16.h>
#include <stdint.h>

// ---------------------------------------------------------------------------
// GaussianAdapter for MI455X (gfx1250).
// Memory-bound streaming kernel (~226 MB traffic -> ~10us floor @ 23.3 TB/s).
//  * CDNA5 TDM: each wave DMAs its 32x19-float raw_gaussians slice into LDS
//    (tensor_load_to_lds + s_wait_tensorcnt).
//  * CDNA5 WMMA: the two per-view constant 9x9 transforms (Rc(x)Rc for the
//    covariance, I3(x)D1 for SH rotation) run on v_wmma_f32_16x16x4_f32.
//  * Hot kernel is branch-free (requires rpix % 256 == 0, true for 352x352);
//    a generic VALU kernel covers any other shape.
// ---------------------------------------------------------------------------

typedef __attribute__((ext_vector_type(4))) unsigned int v4u;
typedef __attribute__((ext_vector_type(4))) int          v4i;
typedef __attribute__((ext_vector_type(8))) int          v8i;
typedef __attribute__((ext_vector_type(2))) float        v2f;
typedef __attribute__((ext_vector_type(8))) float        v8f;

#define TPB 256
#define RAW_PER_PIX 19
#define CST_PER_VIEW 34   // Rc(9) t(3) Kinv(9) D1(9) sh_mask(4)
#define INV_C0 3.5449077018110318f   // 1 / 0.28209479177387814

// ---------------------------------------------------------------------------
// Per-view constant precompute: Kinv, D1, copy Rc/t/sh_mask.  8 views total.
// ---------------------------------------------------------------------------
__global__ void precompute_views(const float* __restrict__ extr,
                                 const float* __restrict__ intr,
                                 const float* __restrict__ sh_mask,
                                 float* __restrict__ cst, int nViews) {
  int v = blockIdx.x * blockDim.x + threadIdx.x;
  if (v >= nViews) return;
  const float* E = extr + v * 16;   // 4x4 row-major
  const float* K = intr + v * 9;    // 3x3 row-major
  float* o = cst + v * CST_PER_VIEW;

  float R[9], t[3];
  #pragma unroll
  for (int i = 0; i < 3; ++i) {
    #pragma unroll
    for (int j = 0; j < 3; ++j) R[i * 3 + j] = E[i * 4 + j];
    t[i] = E[i * 4 + 3];
  }

  float a = K[0], b = K[1], c = K[2];
  float d = K[3], e = K[4], f = K[5];
  float g = K[6], h = K[7], i2 = K[8];
  float A = e * i2 - f * h;
  float B = -(d * i2 - f * g);
  float C = d * h - e * g;
  float inv = 1.0f / (a * A + b * B + c * C);
  float Ki[9];
  Ki[0] = A * inv;               Ki[1] = -(b * i2 - c * h) * inv; Ki[2] = (b * f - c * e) * inv;
  Ki[3] = B * inv;               Ki[4] =  (a * i2 - c * g) * inv; Ki[5] = -(a * f - c * d) * inv;
  Ki[6] = C * inv;               Ki[7] = -(a * h - b * g) * inv;  Ki[8] = (a * e - b * d) * inv;

  const int perm[3] = {1, 2, 0};
  float D1[9];
  #pragma unroll
  for (int i = 0; i < 3; ++i)
    #pragma unroll
    for (int j = 0; j < 3; ++j) D1[i * 3 + j] = R[perm[i] * 3 + perm[j]];

  #pragma unroll
  for (int k = 0; k < 9; ++k) o[k]      = R[k];
  #pragma unroll
  for (int k = 0; k < 3; ++k) o[9 + k]  = t[k];
  #pragma unroll
  for (int k = 0; k < 9; ++k) o[12 + k] = Ki[k];
  #pragma unroll
  for (int k = 0; k < 9; ++k) o[21 + k] = D1[k];
  #pragma unroll
  for (int k = 0; k < 4; ++k) o[30 + k] = sh_mask[k];
}

__device__ __forceinline__ float softplus_clamped(float x) {
  float sp = (x > 0.0f) ? (x + log1pf(expf(-x))) : log1pf(expf(x));
  return fminf(fmaxf(sp, 0.5f), 15.0f);
}

// ---------------------------------------------------------------------------
// Hot kernel: one pixel per thread, branch-free; requires rpix % TPB == 0.
// ---------------------------------------------------------------------------
__global__ void __launch_bounds__(TPB)
gaussian_adapter(const float* __restrict__ cst,
                 const float* __restrict__ coords,
                 const float* __restrict__ depths,
                 const float* __restrict__ opac,
                 const float* __restrict__ raw,
                 const float* __restrict__ imgs,
                 float* __restrict__ out,
                 int N, int rpix) {
  __shared__ float tile[TPB * RAW_PER_PIX];  // TDM destination (first LDS alloc)
  __shared__ float abuf[TPB * 12];           // WMMA A-fragment staging (9 + 3 pad)
  __shared__ float vc[CST_PER_VIEW];

  const int tid  = threadIdx.x;
  const int lane = tid & 31;
  const int wloc = tid & ~31;              // wave base within block
  const int gid0 = blockIdx.x * TPB;
  const int gid  = gid0 + tid;
  const int view = gid0 / rpix;            // uniform per block (rpix % TPB == 0)

  // ---- TDM: each wave async-DMAs its own 32x19-float slice into LDS ----
  {
    const unsigned waveId = (unsigned)wloc >> 5;         // wave-uniform
    const unsigned elems  = 32u * RAW_PER_PIX;           // 608 f32 per wave
    unsigned long long gaddr =
        (unsigned long long)(uintptr_t)raw +
        (unsigned long long)(gid0 + (int)waveId * 32) * (RAW_PER_PIX * 4u);
    unsigned lds_off = (unsigned)(uintptr_t)&tile[0] + waveId * (elems * 4u);

    v4u g0;                                              // D# group 0
    g0.x = 1u;                                           // count=1
    g0.y = lds_off;
    g0.z = (unsigned)(gaddr & 0xffffffffull);
    g0.w = (unsigned)((gaddr >> 32) & 0x01ffffffull) | (2u << 30);   // type=2

    v8i g1;                                              // D# group 1
    g1[0] = (int)(2u << 16);                             // data_size=2 (4B)
    g1[1] = (int)((elems & 0xffffu) << 16);              // tensor_dim0 lo16
    g1[2] = (int)(((elems >> 16) & 0xffffu) | (1u << 16)); // dim0 hi | tensor_dim1=1
    g1[3] = (int)((elems & 0xffffu) << 16);              // tile_dim0
    g1[4] = 1;                                           // tile_dim1=1, tile_dim2=0
    g1[5] = (int)elems;                                  // tensor_dim0_stride lo32
    g1[6] = 0;
    g1[7] = 0;

    v4i gz = {0, 0, 0, 0};
#if defined(__clang_major__) && (__clang_major__ >= 23)
    v8i gz8 = {0, 0, 0, 0, 0, 0, 0, 0};
    __builtin_amdgcn_tensor_load_to_lds(g0, g1, gz, gz, gz8, 0);
#else
    __builtin_amdgcn_tensor_load_to_lds(g0, g1, gz, gz, 0);
#endif
    __builtin_amdgcn_s_wait_tensorcnt(0);
  }

  if (tid < CST_PER_VIEW) vc[tid] = cst[view * CST_PER_VIEW + tid];
  __syncthreads();

  const float* rg = &tile[tid * RAW_PER_PIX]; // stride 19: gcd(19,64)=1, conflict-free
  const float* Rc = &vc[0];
  const float* tr = &vc[9];
  const float* Ki = &vc[12];
  const float* mk = &vc[30];

  // ---- scales: clip(softplus(x-4), 0.5, 15) ----
  float sc0 = softplus_clamped(rg[0] - 4.0f);
  float sc1 = softplus_clamped(rg[1] - 4.0f);
  float sc2 = softplus_clamped(rg[2] - 4.0f);

  // ---- rotation quaternion: q / (|q| + eps) ----
  float qw = rg[3], qx = rg[4], qy = rg[5], qz = rg[6];
  float qn = sqrtf(qw * qw + qx * qx + qy * qy + qz * qz);
  float qi = 1.0f / (qn + 1e-8f);
  qw *= qi; qx *= qi; qy *= qi; qz *= qi;

  // ---- SH: raw[7:19].reshape(3,4) * mask, DC += (img-0.5)/C0 ----
  float sh[3][4];
  #pragma unroll
  for (int c = 0; c < 3; ++c)
    #pragma unroll
    for (int k = 0; k < 4; ++k) sh[c][k] = rg[7 + c * 4 + k] * mk[k];

  const int p = gid - view * rpix;
  const float* ib = imgs + (size_t)view * 3u * (size_t)rpix + (size_t)p;
  #pragma unroll
  for (int c = 0; c < 3; ++c)
    sh[c][0] += (__builtin_nontemporal_load(&ib[(size_t)c * (size_t)rpix]) - 0.5f) * INV_C0;

  // ---- quat -> rotation matrix (two_s = 2/|q|^2, as in reference) ----
  float two_s = 2.0f / (qw * qw + qx * qx + qy * qy + qz * qz);
  float Rq[9];
  Rq[0] = 1.0f - two_s * (qy * qy + qz * qz);
  Rq[1] = two_s * (qx * qy - qz * qw);
  Rq[2] = two_s * (qx * qz + qy * qw);
  Rq[3] = two_s * (qx * qy + qz * qw);
  Rq[4] = 1.0f - two_s * (qx * qx + qz * qz);
  Rq[5] = two_s * (qy * qz - qx * qw);
  Rq[6] = two_s * (qx * qz - qy * qw);
  Rq[7] = two_s * (qy * qz + qx * qw);
  Rq[8] = 1.0f - two_s * (qx * qx + qy * qy);

  // ---- cov0 = Rq diag(s^2) Rq^T (per-pixel; WMMA does Rc cov0 Rc^T) ----
  float s2[3] = {sc0 * sc0, sc1 * sc1, sc2 * sc2};
  float cv0[9];
  #pragma unroll
  for (int i = 0; i < 3; ++i)
    #pragma unroll
    for (int k = 0; k < 3; ++k)
      cv0[i * 3 + k] = Rq[i * 3 + 0] * s2[0] * Rq[k * 3 + 0] +
                       Rq[i * 3 + 1] * s2[1] * Rq[k * 3 + 1] +
                       Rq[i * 3 + 2] * s2[2] * Rq[k * 3 + 2];

  // ---- means = t + Rc * normalize(Kinv [x,y,1]) * depth ----
  float cx  = __builtin_nontemporal_load(&coords[(size_t)gid * 2 + 0]);
  float cyv = __builtin_nontemporal_load(&coords[(size_t)gid * 2 + 1]);
  float dx = Ki[0] * cx + Ki[1] * cyv + Ki[2];
  float dy = Ki[3] * cx + Ki[4] * cyv + Ki[5];
  float dz = Ki[6] * cx + Ki[7] * cyv + Ki[8];
  float dn = 1.0f / sqrtf(dx * dx + dy * dy + dz * dz);
  dx *= dn; dy *= dn; dz *= dn;
  float wx = Rc[0] * dx + Rc[1] * dy + Rc[2] * dz;
  float wy = Rc[3] * dx + Rc[4] * dy + Rc[5] * dz;
  float wz = Rc[6] * dx + Rc[7] * dy + Rc[8] * dz;
  float dep = __builtin_nontemporal_load(&depths[gid]);
  float m0 = tr[0] + wx * dep;
  float m1 = tr[1] + wy * dep;
  float m2 = tr[2] + wz * dep;

  // ------------------------------------------------------------------------
  // WMMA section: v_wmma_f32_16x16x4_f32, full EXEC required (no divergence
  // reaches here in the hot kernel).
  //   cov_flat  = (Rc (x) Rc)  * cv0_flat   (9x9, padded to K=12, N=16)
  //   shrot     = (I3 (x) D1)  * shv        (9x9, block-diagonal)
  // ------------------------------------------------------------------------
  const size_t Ns = (size_t)N;
  float* o_cov  = out + 3 * Ns;
  float* o_harm = out + 19 * Ns;

  const int  ncol   = lane & 15;
  const bool nvalid = (ncol < 9);
  const int  khalf  = (lane < 16) ? 0 : 2;
  const int  m_add  = (lane < 16) ? 0 : 8;

  auto t9 = [&](int j) -> float {   // T9[n][j] = Rc[n/3][j/3] * Rc[n%3][j%3]
    if (!nvalid || j >= 9) return 0.0f;
    return vc[(ncol / 3) * 3 + (j / 3)] * vc[(ncol % 3) * 3 + (j % 3)];
  };
  auto h9 = [&](int j) -> float {   // H9[n][j] = (j/3==n/3) ? D1[n%3][j%3] : 0
    if (!nvalid || j >= 9) return 0.0f;
    return ((j / 3) == (ncol / 3)) ? vc[21 + (ncol % 3) * 3 + (j % 3)] : 0.0f;
  };

  v2f b_cov[3], b_sh[3];
  #pragma unroll
  for (int c = 0; c < 3; ++c) {
    int j0 = c * 4 + khalf;
    b_cov[c].x = t9(j0);     b_cov[c].y = t9(j0 + 1);
    b_sh[c].x  = h9(j0);     b_sh[c].y  = h9(j0 + 1);
  }

  // --- phase 1: covariance ---
  #pragma unroll
  for (int k = 0; k < 9; ++k) abuf[tid * 12 + k] = cv0[k];
  abuf[tid * 12 + 9] = 0.0f; abuf[tid * 12 + 10] = 0.0f; abuf[tid * 12 + 11] = 0.0f;
  __builtin_amdgcn_wave_barrier();   // intra-wave only: LDS is in-order per wave

  #pragma unroll
  for (int g = 0; g < 2; ++g) {
    const int arow = wloc + g * 16 + ncol;
    v8f acc = {};
    #pragma unroll
    for (int c = 0; c < 3; ++c) {
      v2f a;
      a.x = abuf[arow * 12 + c * 4 + khalf];
      a.y = abuf[arow * 12 + c * 4 + khalf + 1];
      acc = __builtin_amdgcn_wmma_f32_16x16x4_f32(false, a, false, b_cov[c],
                                                  (short)0, acc, false, false);
    }
    if (nvalid) {
      const int pixb = gid0 + wloc + g * 16 + m_add;
      float* bp = o_cov + (size_t)pixb * 9 + ncol;   // one base, immediate offsets
      #pragma unroll
      for (int r = 0; r < 8; ++r) bp[r * 9] = acc[r];
    }
  }

  // --- phase 2: SH degree-1 rotation ---
  __builtin_amdgcn_wave_barrier();
  #pragma unroll
  for (int c = 0; c < 3; ++c)
    #pragma unroll
    for (int jj = 0; jj < 3; ++jj) abuf[tid * 12 + c * 3 + jj] = sh[c][1 + jj];
  abuf[tid * 12 + 9] = 0.0f; abuf[tid * 12 + 10] = 0.0f; abuf[tid * 12 + 11] = 0.0f;
  __builtin_amdgcn_wave_barrier();

  #pragma unroll
  for (int g = 0; g < 2; ++g) {
    const int arow = wloc + g * 16 + ncol;
    v8f acc = {};
    #pragma unroll
    for (int c = 0; c < 3; ++c) {
      v2f a;
      a.x = abuf[arow * 12 + c * 4 + khalf];
      a.y = abuf[arow * 12 + c * 4 + khalf + 1];
      acc = __builtin_amdgcn_wmma_f32_16x16x4_f32(false, a, false, b_sh[c],
                                                  (short)0, acc, false, false);
    }
    if (nvalid) {
      const int hoff = (ncol / 3) * 4 + 1 + (ncol % 3);   // harm[c][1+i] slot
      const int pixb = gid0 + wloc + g * 16 + m_add;
      float* bp = o_harm + (size_t)pixb * 12 + hoff;
      #pragma unroll
      for (int r = 0; r < 8; ++r) bp[r * 12] = acc[r];
    }
  }

  // ---- per-thread stores for the remaining outputs (non-temporal) ----
  {
    float* o_means = out;
    float* o_scl   = out + 12 * Ns;
    float* o_rot   = out + 15 * Ns;
    float* o_op    = out + 31 * Ns;
    size_t g = (size_t)gid;
    __builtin_nontemporal_store(m0, &o_means[g * 3 + 0]);
    __builtin_nontemporal_store(m1, &o_means[g * 3 + 1]);
    __builtin_nontemporal_store(m2, &o_means[g * 3 + 2]);
    __builtin_nontemporal_store(sc0, &o_scl[g * 3 + 0]);
    __builtin_nontemporal_store(sc1, &o_scl[g * 3 + 1]);
    __builtin_nontemporal_store(sc2, &o_scl[g * 3 + 2]);
    __builtin_nontemporal_store(qw, &o_rot[g * 4 + 0]);
    __builtin_nontemporal_store(qx, &o_rot[g * 4 + 1]);
    __builtin_nontemporal_store(qy, &o_rot[g * 4 + 2]);
    __builtin_nontemporal_store(qz, &o_rot[g * 4 + 3]);
    #pragma unroll
    for (int c = 0; c < 3; ++c) o_harm[g * 12 + c * 4] = sh[c][0];   // DC terms
    __builtin_nontemporal_store(__builtin_nontemporal_load(&opac[gid]), &o_op[g]);
  }
}

// ---------------------------------------------------------------------------
// Generic fallback (any shape): plain per-thread VALU version.
// ---------------------------------------------------------------------------
__global__ void __launch_bounds__(TPB)
gaussian_adapter_generic(const float* __restrict__ cst,
                         const float* __restrict__ coords,
                         const float* __restrict__ depths,
                         const float* __restrict__ opac,
                         const float* __restrict__ raw,
                         const float* __restrict__ imgs,
                         float* __restrict__ out,
                         int N, int rpix) {
  const int gid = blockIdx.x * TPB + threadIdx.x;
  if (gid >= N) return;
  const int view = gid / rpix;
  const int p    = gid - view * rpix;
  const float* vcg = cst + view * CST_PER_VIEW;
  float vcl[CST_PER_VIEW];
  #pragma unroll
  for (int k = 0; k < CST_PER_VIEW; ++k) vcl[k] = vcg[k];
  const float* Rc = &vcl[0];
  const float* tr = &vcl[9];
  const float* Ki = &vcl[12];
  const float* D1 = &vcl[21];
  const float* mk = &vcl[30];

  float rg[RAW_PER_PIX];
  #pragma unroll
  for (int k = 0; k < RAW_PER_PIX; ++k)
    rg[k] = raw[(size_t)gid * RAW_PER_PIX + k];

  float sc0 = softplus_clamped(rg[0] - 4.0f);
  float sc1 = softplus_clamped(rg[1] - 4.0f);
  float sc2 = softplus_clamped(rg[2] - 4.0f);

  float qw = rg[3], qx = rg[4], qy = rg[5], qz = rg[6];
  float qn = sqrtf(qw * qw + qx * qx + qy * qy + qz * qz);
  float qi = 1.0f / (qn + 1e-8f);
  qw *= qi; qx *= qi; qy *= qi; qz *= qi;

  float sh[3][4];
  #pragma unroll
  for (int c = 0; c < 3; ++c)
    #pragma unroll
    for (int k = 0; k < 4; ++k) sh[c][k] = rg[7 + c * 4 + k] * mk[k];
  const float* ib = imgs + (size_t)view * 3u * (size_t)rpix + (size_t)p;
  #pragma unroll
  for (int c = 0; c < 3; ++c)
    sh[c][0] += (ib[(size_t)c * (size_t)rpix] - 0.5f) * INV_C0;

  float two_s = 2.0f / (qw * qw + qx * qx + qy * qy + qz * qz);
  float Rq[9];
  Rq[0] = 1.0f - two_s * (qy * qy + qz * qz);
  Rq[1] = two_s * (qx * qy - qz * qw);
  Rq[2] = two_s * (qx * qz + qy * qw);
  Rq[3] = two_s * (qx * qy + qz * qw);
  Rq[4] = 1.0f - two_s * (qx * qx + qz * qz);
  Rq[5] = two_s * (qy * qz - qx * qw);
  Rq[6] = two_s * (qx * qz - qy * qw);
  Rq[7] = two_s * (qy * qz + qx * qw);
  Rq[8] = 1.0f - two_s * (qx * qx + qy * qy);

  float s2[3] = {sc0 * sc0, sc1 * sc1, sc2 * sc2};
  float cv0[9], tm[9], cov[9];
  #pragma unroll
  for (int i = 0; i < 3; ++i)
    #pragma unroll
    for (int k = 0; k < 3; ++k)
      cv0[i * 3 + k] = Rq[i * 3 + 0] * s2[0] * Rq[k * 3 + 0] +
                       Rq[i * 3 + 1] * s2[1] * Rq[k * 3 + 1] +
                       Rq[i * 3 + 2] * s2[2] * Rq[k * 3 + 2];
  #pragma unroll
  for (int i = 0; i < 3; ++i)
    #pragma unroll
    for (int k = 0; k < 3; ++k)
      tm[i * 3 + k] = Rc[i * 3] * cv0[k] + Rc[i * 3 + 1] * cv0[3 + k] + Rc[i * 3 + 2] * cv0[6 + k];
  #pragma unroll
  for (int i = 0; i < 3; ++i)
    #pragma unroll
    for (int l = 0; l < 3; ++l)
      cov[i * 3 + l] = tm[i * 3] * Rc[l * 3] + tm[i * 3 + 1] * Rc[l * 3 + 1] + tm[i * 3 + 2] * Rc[l * 3 + 2];

  float cx  = coords[(size_t)gid * 2 + 0];
  float cyv = coords[(size_t)gid * 2 + 1];
  float dx = Ki[0] * cx + Ki[1] * cyv + Ki[2];
  float dy = Ki[3] * cx + Ki[4] * cyv + Ki[5];
  float dz = Ki[6] * cx + Ki[7] * cyv + Ki[8];
  float dn = 1.0f / sqrtf(dx * dx + dy * dy + dz * dz);
  dx *= dn; dy *= dn; dz *= dn;
  float wxv = Rc[0] * dx + Rc[1] * dy + Rc[2] * dz;
  float wyv = Rc[3] * dx + Rc[4] * dy + Rc[5] * dz;
  float wzv = Rc[6] * dx + Rc[7] * dy + Rc[8] * dz;
  float dep = depths[gid];

  float harm[3][4];
  #pragma unroll
  for (int c = 0; c < 3; ++c) {
    harm[c][0] = sh[c][0];
    #pragma unroll
    for (int i = 0; i < 3; ++i)
      harm[c][1 + i] = D1[i * 3] * sh[c][1] + D1[i * 3 + 1] * sh[c][2] + D1[i * 3 + 2] * sh[c][3];
  }

  const size_t Ns = (size_t)N;
  size_t g = (size_t)gid;
  float* o_means = out;
  float* o_cov   = out + 3 * Ns;
  float* o_scl   = out + 12 * Ns;
  float* o_rot   = out + 15 * Ns;
  float* o_harm  = out + 19 * Ns;
  float* o_op    = out + 31 * Ns;
  o_means[g * 3 + 0] = tr[0] + wxv * dep;
  o_means[g * 3 + 1] = tr[1] + wyv * dep;
  o_means[g * 3 + 2] = tr[2] + wzv * dep;
  #pragma unroll
  for (int k = 0; k < 9; ++k) o_cov[g * 9 + k] = cov[k];
  o_scl[g * 3 + 0] = sc0;
  o_scl[g * 3 + 1] = sc1;
  o_scl[g * 3 + 2] = sc2;
  o_rot[g * 4 + 0] = qw;
  o_rot[g * 4 + 1] = qx;
  o_rot[g * 4 + 2] = qy;
  o_rot[g * 4 + 3] = qz;
  #pragma unroll
  for (int c = 0; c < 3; ++c)
    #pragma unroll
    for (int k = 0; k < 4; ++k) o_harm[g * 12 + c * 4 + k] = harm[c][k];
  o_op[g] = opac[gid];
}

// ---------------------------------------------------------------------------
extern "C" void kernel_launch(void* const* d_in, const int* in_sizes, int n_in,
                              void* d_out, int out_size, void* d_ws, size_t ws_size,
                              hipStream_t stream) {
  const float* extr   = (const float*)d_in[0];
  const float* intr   = (const float*)d_in[1];
  const float* coords = (const float*)d_in[2];
  const float* depths = (const float*)d_in[3];
  const float* opacs  = (const float*)d_in[4];
  const float* raw    = (const float*)d_in[5];
  const float* imgs   = (const float*)d_in[6];
  const float* shm    = (const float*)d_in[7];

  const int nViews = in_sizes[0] / 16;   // extrinsics = nViews * 4*4
  const int N      = in_sizes[3];        // depths flat count = B*V*R_PIX
  const int rpix   = N / nViews;

  float* cst = (float*)d_ws;             // nViews * 34 floats of scratch

  precompute_views<<<1, 32, 0, stream>>>(extr, intr, shm, cst, nViews);

  if ((rpix % TPB) == 0) {
    // hot path: blocks never straddle views, N % TPB == 0 -> branch-free kernel
    gaussian_adapter<<<N / TPB, TPB, 0, stream>>>(cst, coords, depths, opacs, raw,
                                                  imgs, (float*)d_out, N, rpix);
  } else {
    gaussian_adapter_generic<<<(N + TPB - 1) / TPB, TPB, 0, stream>>>(
        cst, coords, depths, opacs, raw, imgs, (float*)d_out, N, rpix);
  }
}